// CapsuleLayer_30992484008534
// MI455X (gfx1250) — compile-verified
//
#include <hip/hip_runtime.h>

typedef __attribute__((ext_vector_type(2))) float v2f;
typedef __attribute__((ext_vector_type(8))) float v8f;

constexpr int BATCH = 8192;
constexpr int LMAX  = 50;
constexpr int NU    = 64;          // IN_U == OUT_U
constexpr int KCAP  = 4;
constexpr int NITER = 3;
constexpr int RB    = 8;           // batch rows per block
constexpr int MROWS = RB * LMAX;   // 400 (b,l) rows per block tile
constexpr int MT    = MROWS / 16;  // 25 m-tiles per block
constexpr int EST   = 65;          // padded LDS row stride (floats) -> conflict-free delta phase
constexpr int WPS   = 80;          // float2 stride per k-pair row of W: 160 dwords === 32 (mod 64)
                                   // -> lanes 0-15 and 16-31 of a b64 load use disjoint bank halves

static __device__ __forceinline__ v8f wmma_f32(v2f a, v2f b, v8f c) {
  // D(16x16 f32) = A(16x4 f32) x B(4x16 f32) + C
  return __builtin_amdgcn_wmma_f32_16x16x4_f32(false, a, false, b, (short)0, c, false, false);
}

// Stage W into pair-packed LDS: Wp[(k>>1)*WPS + col] = {W[k][col], W[k+1][col]}
// so each WMMA B fragment is ONE aligned ds_load_b64 into an even VGPR pair.
static __device__ __forceinline__ void stage_W(const float* __restrict__ W,
                                               float* __restrict__ Wpf, int tid) {
  for (int i = tid; i < NU * NU; i += 256) {
    const int k = i >> 6, c = i & 63;
    Wpf[((k >> 1) * WPS + c) * 2 + (k & 1)] = W[i];
  }
}

// ---------------------------------------------------------------------------
// Kernel: precompute emb_map = behavior @ W into workspace (used if ws fits).
// Grid: (BATCH*LMAX/16)/8 blocks of 256 threads; each wave owns one 16-row M-tile.
// ---------------------------------------------------------------------------
__global__ __launch_bounds__(256)
void caps_embmap_kernel(const float* __restrict__ behavior,
                        const float* __restrict__ W,
                        float* __restrict__ out) {
  __shared__ alignas(16) float Wpf[32 * WPS * 2];
  const int tid = threadIdx.x;
  stage_W(W, Wpf, tid);
  __syncthreads();
  const v2f* Wp = (const v2f*)Wpf;

  const int wave = tid >> 5;
  const int lane = tid & 31;
  const int lh   = lane >> 4;   // lane half: 0 -> K+{0,1}, 1 -> K+{2,3}
  const int lm   = lane & 15;

  const int mt = blockIdx.x * 8 + wave;          // global m-tile, 25600 total
  v8f acc[4] = {};
  const float* arow = behavior + ((size_t)mt * 16 + lm) * NU + lh * 2;

  for (int s = 0; s < 16; ++s) {
    v2f a = { arow[4 * s], arow[4 * s + 1] };
    const int pr = 2 * s + lh;                   // k-pair row = (4s + 2*lh) >> 1
#pragma unroll
    for (int n = 0; n < 4; ++n)
      acc[n] = wmma_f32(a, Wp[pr * WPS + n * 16 + lm], acc[n]);
  }
#pragma unroll
  for (int n = 0; n < 4; ++n)
#pragma unroll
    for (int v = 0; v < 8; ++v)
      out[((size_t)mt * 16 + v + lh * 8) * NU + n * 16 + lm] = acc[n][v];
}

// ---------------------------------------------------------------------------
// Main routing-iteration kernel. One block = 8 batch rows.
// Phases: stage emb_map tile in LDS (WMMA or copy from ws) -> masked softmax
//         -> Z + squash (caps) -> optional global delta atomic-reduction.
// ---------------------------------------------------------------------------
__global__ __launch_bounds__(256)
void caps_iter_kernel(const float* __restrict__ behavior,   // [B,L,64]
                      const int*   __restrict__ seq_len,    // [B]
                      const float* __restrict__ W,          // [64,64]
                      const float* __restrict__ logits,     // [4*50]
                      float*       __restrict__ delta,      // [4*50]
                      const float* __restrict__ emb_ws,     // nullptr or [B*L,64]
                      float*       __restrict__ caps_out,   // nullptr or [B,4,64]
                      int do_delta) {
  __shared__ float emb_lds[MROWS * EST];           // 400x64 (stride 65) ~104 KB
  __shared__ alignas(16) float Wpf[32 * WPS * 2];  // pair-packed W, 20.5 KB
  __shared__ float lg[KCAP * LMAX];                // 200
  __shared__ float w_lds[RB * KCAP * LMAX];        // 1600
  __shared__ float caps_lds[RB * KCAP * EST];      // 32x64 (stride 65)
  __shared__ int   slen[RB];

  const int tid  = threadIdx.x;
  const int b0   = blockIdx.x * RB;
  const int lane = tid & 31;
  const int wave = tid >> 5;

  // ---- phase 0: stage small constants -------------------------------------
  for (int i = tid; i < KCAP * LMAX; i += 256) lg[i] = logits[i];
  if (tid < RB) slen[tid] = seq_len[b0 + tid];
  if (emb_ws == nullptr) stage_W(W, Wpf, tid);
  __syncthreads();

  // ---- phase 1: emb_map tile into LDS -------------------------------------
  if (emb_ws != nullptr) {
    const float* src = emb_ws + (size_t)b0 * LMAX * NU;
    for (int i = tid; i < MROWS * 16; i += 256) {
      const int r = i >> 4, c = (i & 15) * 4;
      const float4 v = *(const float4*)(src + (size_t)r * NU + c);
      float* d = &emb_lds[r * EST + c];
      d[0] = v.x; d[1] = v.y; d[2] = v.z; d[3] = v.w;
    }
  } else {
    const v2f* Wp = (const v2f*)Wpf;
    const int lh = lane >> 4, lm = lane & 15;
    for (int mt = wave; mt < MT; mt += 8) {
      v8f acc[4] = {};
      const float* arow = behavior + ((size_t)(b0 * LMAX + mt * 16 + lm)) * NU + lh * 2;
      if (mt + 8 < MT)  // prefetch next m-tile's A row (global_prefetch_b8)
        __builtin_prefetch(arow + (size_t)8 * 16 * NU, 0, 0);
      for (int s = 0; s < 16; ++s) {
        v2f a = { arow[4 * s], arow[4 * s + 1] };
        const int pr = 2 * s + lh;
#pragma unroll
        for (int n = 0; n < 4; ++n)
          acc[n] = wmma_f32(a, Wp[pr * WPS + n * 16 + lm], acc[n]);
      }
#pragma unroll
      for (int n = 0; n < 4; ++n)
#pragma unroll
        for (int v = 0; v < 8; ++v)
          emb_lds[(mt * 16 + v + lh * 8) * EST + n * 16 + lm] = acc[n][v];
    }
  }
  __syncthreads();

  // ---- phase 2: masked softmax over L (mask is per-batch-row) -------------
  {
    const int p   = tid >> 3;          // (b_local,k) pair 0..31
    const int sub = tid & 7;           // 8 threads per pair
    const int bl  = p >> 2, k = p & 3;
    const int sl  = slen[bl];
    float mx = -3.0e38f;
    for (int l = sub; l < LMAX; l += 8)
      if (l < sl) mx = fmaxf(mx, lg[k * LMAX + l]);
#pragma unroll
    for (int off = 1; off < 8; off <<= 1) mx = fmaxf(mx, __shfl_xor(mx, off, 32));
    float sum = 0.0f;
    for (int l = sub; l < LMAX; l += 8) {
      const float e = (l < sl) ? __expf(lg[k * LMAX + l] - mx) : 0.0f;
      w_lds[p * LMAX + l] = e;
      sum += e;
    }
#pragma unroll
    for (int off = 1; off < 8; off <<= 1) sum += __shfl_xor(sum, off, 32);
    const float inv = 1.0f / sum;
    for (int l = sub; l < LMAX; l += 8) w_lds[p * LMAX + l] *= inv;
  }
  __syncthreads();

  // ---- phase 3: Z = w @ emb_map, squash -> caps ---------------------------
  for (int p = wave; p < RB * KCAP; p += 8) {     // one wave per (b,k) pair
    const int bl = p >> 2, k = p & 3;
    const int o0 = lane, o1 = lane + 32;
    float z0 = 0.0f, z1 = 0.0f;
    const float* wr = &w_lds[p * LMAX];
    const float* er = &emb_lds[bl * LMAX * EST];
    for (int l = 0; l < LMAX; ++l) {
      const float wl = wr[l];
      z0 = fmaf(wl, er[l * EST + o0], z0);
      z1 = fmaf(wl, er[l * EST + o1], z1);
    }
    float n2 = z0 * z0 + z1 * z1;
#pragma unroll
    for (int off = 1; off < 32; off <<= 1) n2 += __shfl_xor(n2, off, 32);
    const float scale = n2 / ((1.0f + n2) * sqrtf(n2 + 1e-8f));
    const float c0 = scale * z0, c1 = scale * z1;
    caps_lds[p * EST + o0] = c0;
    caps_lds[p * EST + o1] = c1;
    if (caps_out != nullptr) {
      float* dst = caps_out + (((size_t)(b0 + bl)) * KCAP + k) * NU;
      dst[o0] = c0;
      dst[o1] = c1;
    }
  }
  __syncthreads();

  // ---- phase 4: delta[k,l] += sum_{b in tile} caps[b,k,:] . emb_map[b,l,:] -
  if (do_delta) {
    for (int t = tid; t < KCAP * LMAX; t += 256) {
      const int k = t / LMAX, l = t % LMAX;
      float acc = 0.0f;
      for (int bl = 0; bl < RB; ++bl) {
        const float* c = &caps_lds[(bl * KCAP + k) * EST];
        const float* e = &emb_lds[(bl * LMAX + l) * EST];
#pragma unroll 8
        for (int o = 0; o < NU; ++o) acc = fmaf(c[o], e[o], acc);
      }
      atomicAdd(&delta[t], acc);
    }
  }
}

// ---------------------------------------------------------------------------
__global__ void caps_init_kernel(const float* __restrict__ routing,
                                 float* __restrict__ logits,
                                 float* __restrict__ delta) {
  const int t = threadIdx.x;
  if (t < KCAP * LMAX) { logits[t] = routing[t]; delta[t] = 0.0f; }
}

__global__ void caps_update_kernel(float* __restrict__ logits,
                                   float* __restrict__ delta) {
  const int t = threadIdx.x;
  if (t < KCAP * LMAX) { logits[t] += delta[t]; delta[t] = 0.0f; }
}

// ---------------------------------------------------------------------------
extern "C" void kernel_launch(void* const* d_in, const int* in_sizes, int n_in,
                              void* d_out, int out_size, void* d_ws, size_t ws_size,
                              hipStream_t stream) {
  const float* behavior = (const float*)d_in[0];   // [8192,50,64] f32
  const int*   seq_len  = (const int*)  d_in[1];   // [8192,1] i32
  const float* routing  = (const float*)d_in[2];   // [1,4,50] f32
  const float* W        = (const float*)d_in[3];   // [64,64] f32
  float*       out      = (float*)d_out;           // [8192,4,64] f32
  float*       ws       = (float*)d_ws;

  float* logits_buf = ws;          // 200 floats (padded region 256)
  float* delta_buf  = ws + 256;    // 200 floats (padded region 256)
  float* emb_ws     = ws + 512;    // [B*L,64] f32 if it fits

  const size_t need = (512 + (size_t)BATCH * LMAX * NU) * sizeof(float);
  const bool use_ws = ws_size >= need;

  caps_init_kernel<<<1, 256, 0, stream>>>(routing, logits_buf, delta_buf);
  if (use_ws) {
    const int mtiles = BATCH * LMAX / 16;          // 25600
    caps_embmap_kernel<<<mtiles / 8, 256, 0, stream>>>(behavior, W, emb_ws);
  }
  for (int it = 0; it < NITER; ++it) {
    const bool last = (it == NITER - 1);
    caps_iter_kernel<<<BATCH / RB, 256, 0, stream>>>(
        behavior, seq_len, W, logits_buf, delta_buf,
        use_ws ? emb_ws : nullptr,
        last ? out : nullptr,
        last ? 0 : 1);
    if (!last) caps_update_kernel<<<1, 256, 0, stream>>>(logits_buf, delta_buf);
  }
}